// PostProcessor_13666585936350
// MI455X (gfx1250) — compile-verified
//
#include <hip/hip_runtime.h>
#include <math.h>

typedef float v2f __attribute__((ext_vector_type(2)));
typedef float v8f __attribute__((ext_vector_type(8)));

constexpr int   kB       = 2;
constexpr int   kN       = 1024;
constexpr int   kC       = 81;
constexpr int   kFG      = 80;      // foreground classes 1..80
constexpr int   kDets    = 100;
constexpr float kScoreTh = 0.05f;
constexpr float kNmsTh   = 0.5f;
constexpr float kNegInf  = -1.0e30f;
constexpr float kXformClip = 4.135166556742356f;  // log(1000/16)
constexpr float kMaxX    = 1332.0f;               // IMG_W - 1
constexpr float kMaxY    = 799.0f;                // IMG_H - 1

// ---------------------------------------------------------------------------
// Kernel 1: softmax over 81 classes + box decode + clip.  One wave per row.
// ---------------------------------------------------------------------------
__global__ void softmax_decode_kernel(const float* __restrict__ logits,
                                      const float* __restrict__ reg,
                                      const float* __restrict__ props,
                                      float* __restrict__ prob,
                                      float* __restrict__ boxes) {
  const int row  = blockIdx.x * 8 + (threadIdx.x >> 5);
  const int lane = threadIdx.x & 31;
  if (row >= kB * kN) return;

  const float* lg = logits + (size_t)row * kC;
  float m = -3.0e38f;
  for (int c = lane; c < kC; c += 32) m = fmaxf(m, lg[c]);
  for (int o = 16; o > 0; o >>= 1) m = fmaxf(m, __shfl_xor(m, o, 32));

  float* pr = prob + (size_t)row * kC;
  float sum = 0.0f;
  for (int c = lane; c < kC; c += 32) {
    float e = expf(lg[c] - m);
    pr[c] = e;
    sum += e;
  }
  for (int o = 16; o > 0; o >>= 1) sum += __shfl_xor(sum, o, 32);
  const float inv = 1.0f / sum;
  for (int c = lane; c < kC; c += 32) pr[c] *= inv;

  const float px1 = props[row * 4 + 0], py1 = props[row * 4 + 1];
  const float px2 = props[row * 4 + 2], py2 = props[row * 4 + 3];
  const float w = px2 - px1 + 1.0f, h = py2 - py1 + 1.0f;
  const float cx = px1 + 0.5f * w,  cy = py1 + 0.5f * h;

  const float* rr = reg   + (size_t)row * (kC * 4);
  float*       bo = boxes + (size_t)row * (kC * 4);
  for (int c = lane; c < kC; c += 32) {
    float dx = rr[c * 4 + 0] / 10.0f;
    float dy = rr[c * 4 + 1] / 10.0f;
    float dw = fminf(rr[c * 4 + 2] / 5.0f, kXformClip);
    float dh = fminf(rr[c * 4 + 3] / 5.0f, kXformClip);
    float pcx = dx * w + cx, pcy = dy * h + cy;
    float pw = expf(dw) * w, ph = expf(dh) * h;
    float x1 = pcx - 0.5f * pw, y1 = pcy - 0.5f * ph;
    float x2 = pcx + 0.5f * pw - 1.0f, y2 = pcy + 0.5f * ph - 1.0f;
    x1 = fminf(fmaxf(x1, 0.0f), kMaxX);
    x2 = fminf(fmaxf(x2, 0.0f), kMaxX);
    y1 = fminf(fmaxf(y1, 0.0f), kMaxY);
    y2 = fminf(fmaxf(y2, 0.0f), kMaxY);
    bo[c * 4 + 0] = x1; bo[c * 4 + 1] = y1;
    bo[c * 4 + 2] = x2; bo[c * 4 + 3] = y2;
  }
}

// ---------------------------------------------------------------------------
// Kernel 2: one block per (image, fg-class).  LDS-resident bitonic sort +
// WMMA-tiled suppression-bitmask NMS + serial scan.  ~152 KB dynamic LDS.
// ---------------------------------------------------------------------------
__global__ __launch_bounds__(1024) void nms_kernel(const float* __restrict__ prob,
                                                   const float* __restrict__ boxes,
                                                   float* __restrict__ masked) {
  extern __shared__ unsigned char smem[];
  float*    ss    = (float*)(smem);                    // [1024] sorted scores
  int*      sidx  = (int*)(smem + 4096);               // [1024] payload (orig idx)
  float*    bx1   = (float*)(smem + 8192);             // box SoA by orig idx
  float*    by1   = (float*)(smem + 12288);
  float*    bx2   = (float*)(smem + 16384);
  float*    by2   = (float*)(smem + 20480);
  unsigned* maskW = (unsigned*)(smem + 24576);         // [1024][32] suppression bits
  int*      sV    = (int*)(smem + 24576 + 131072);

  const int tid = threadIdx.x;
  const int b   = blockIdx.x / kFG;
  const int cls = blockIdx.x % kFG + 1;

  if (tid == 0) *sV = 0;
  __syncthreads();

  {
    const int n = tid;
    const float p = prob[((size_t)b * kN + n) * kC + cls];
    const bool valid = p > kScoreTh;
    ss[n]   = valid ? p : kNegInf;
    sidx[n] = n;
    const float* bp = boxes + (((size_t)b * kN + n) * kC + cls) * 4;
    bx1[n] = bp[0]; by1[n] = bp[1]; bx2[n] = bp[2]; by2[n] = bp[3];
    masked[((size_t)b * kN + n) * kFG + (cls - 1)] = kNegInf;  // init column
    if (valid) atomicAdd(sV, 1);
  }
  __syncthreads();
  const int V = *sV;

  // Bitonic sort (descending) of 1024 (score, idx) pairs, 1 element/thread.
  for (int k = 2; k <= kN; k <<= 1) {
    for (int j = k >> 1; j > 0; j >>= 1) {
      const int i = tid;
      const int ixj = i ^ j;
      if (ixj > i) {
        const float si = ss[i], sj = ss[ixj];
        const bool up = ((i & k) == 0);
        const bool sw = up ? (si < sj) : (si > sj);
        if (sw) {
          ss[i] = sj; ss[ixj] = si;
          const int t0 = sidx[i]; sidx[i] = sidx[ixj]; sidx[ixj] = t0;
        }
      }
      __syncthreads();
    }
  }

  const int nt = (V + 15) >> 4;  // only V x V block of sorted order matters
  for (int w = tid; w < nt * 16 * 32; w += 1024) maskW[w] = 0u;
  __syncthreads();

  // Phase A: per-wave 16x16 IoU tiles.  area_i + area_j via WMMA rank-2 GEMM.
  const int wave = tid >> 5, lane = tid & 31;
  const int ntt = nt * nt;
  for (int t = wave; t < ntt; t += 32) {
    const int ti = t / nt, tj = t % nt;
    if (tj < ti) continue;  // only j > i can suppress
    const int i0 = ti << 4, j0 = tj << 4;

    // A (16x4 f32): lanes 0-15 hold K=0 (v.x), K=1 (v.y); lanes 16-31 -> K=2,3 = 0
    // B (4x16 f32): lanes 0-15 hold K=0 (v.x), K=1 (v.y) for column n=lane
    v2f av = {0.0f, 0.0f}, bv = {0.0f, 0.0f};
    if (lane < 16) {
      const int oi = sidx[i0 + lane];
      av.x = (bx2[oi] - bx1[oi] + 1.0f) * (by2[oi] - by1[oi] + 1.0f);
      av.y = 1.0f;
      const int oj = sidx[j0 + lane];
      bv.x = 1.0f;
      bv.y = (bx2[oj] - bx1[oj] + 1.0f) * (by2[oj] - by1[oj] + 1.0f);
    }
    v8f acc = {0.f, 0.f, 0.f, 0.f, 0.f, 0.f, 0.f, 0.f};
    acc = __builtin_amdgcn_wmma_f32_16x16x4_f32(false, av, false, bv,
                                                (short)0, acc, false, false);
    // C layout: lanes 0-15: M=v, N=lane; lanes 16-31: M=v+8, N=lane-16
    const int n  = lane & 15;
    const int mb = (lane >> 4) << 3;
    const int J  = j0 + n;
    const int oj = sidx[J];
    const float jx1 = bx1[oj], jy1 = by1[oj], jx2 = bx2[oj], jy2 = by2[oj];
#pragma unroll
    for (int v = 0; v < 8; ++v) {
      const int I  = i0 + mb + v;
      const int oi = sidx[I];
      const float xx1 = fmaxf(bx1[oi], jx1), yy1 = fmaxf(by1[oi], jy1);
      const float xx2 = fminf(bx2[oi], jx2), yy2 = fminf(by2[oi], jy2);
      const float iw = fmaxf(xx2 - xx1 + 1.0f, 0.0f);
      const float ih = fmaxf(yy2 - yy1 + 1.0f, 0.0f);
      const float inter = iw * ih;
      const float iou = inter / (acc[v] - inter);  // acc = area_i + area_j
      if ((J > I) && (J < V) && (I < V) && (iou > kNmsTh))
        atomicOr(&maskW[(I << 5) + (J >> 5)], 1u << (J & 31));
    }
  }
  __syncthreads();

  // Phase B: serial greedy scan over at most V sorted candidates.
  if (tid == 0) {
    unsigned rem[32];
#pragma unroll
    for (int w = 0; w < 32; ++w) rem[w] = 0u;
    for (int i = 0; i < V; ++i) {
      if (!((rem[i >> 5] >> (i & 31)) & 1u)) {
        const int orig = sidx[i];
        masked[((size_t)b * kN + orig) * kFG + (cls - 1)] = ss[i];  // kept
        const unsigned* R = &maskW[i << 5];
        for (int w = 0; w < 32; ++w) rem[w] |= R[w];
      }
    }
  }
}

// ---------------------------------------------------------------------------
// Kernel 3: per-image iterative top-100 (argmax with lowest-index tie-break,
// matching jax.lax.top_k).  One block per image.
// ---------------------------------------------------------------------------
__global__ __launch_bounds__(1024) void topk_kernel(float* __restrict__ masked,
                                                    const float* __restrict__ boxes,
                                                    float* __restrict__ out) {
  __shared__ float rv[1024];
  __shared__ int   ri[1024];
  const int b = blockIdx.x, tid = threadIdx.x;
  float* ms = masked + (size_t)b * (kN * kFG);
  float* obox = out;                                   // [B][100][4]
  float* osc  = out + (size_t)kB * kDets * 4;          // [B][100]
  float* olab = out + (size_t)kB * kDets * 4 + (size_t)kB * kDets;  // [B][100]
  const float kRemoved = -2.0e30f;

  for (int it = 0; it < kDets; ++it) {
    float bvv = kRemoved; int bii = 0x7fffffff;
    for (int f = tid; f < kN * kFG; f += 1024) {
      const float v = ms[f];
      if (v > bvv || (v == bvv && f < bii)) { bvv = v; bii = f; }
    }
    rv[tid] = bvv; ri[tid] = bii;
    __syncthreads();
    for (int s = 512; s > 0; s >>= 1) {
      if (tid < s) {
        const float ov = rv[tid + s]; const int oi = ri[tid + s];
        if (ov > rv[tid] || (ov == rv[tid] && oi < ri[tid])) { rv[tid] = ov; ri[tid] = oi; }
      }
      __syncthreads();
    }
    if (tid == 0) {
      const float sv = rv[0]; const int fi = ri[0];
      const bool ok = sv > kScoreTh;
      const int n = fi / kFG, cl = fi % kFG + 1;
      const float* bp = boxes + (((size_t)b * kN + n) * kC + cl) * 4;
      float* ob = obox + ((size_t)b * kDets + it) * 4;
      ob[0] = ok ? bp[0] : 0.0f;
      ob[1] = ok ? bp[1] : 0.0f;
      ob[2] = ok ? bp[2] : 0.0f;
      ob[3] = ok ? bp[3] : 0.0f;
      osc[b * kDets + it]  = ok ? sv : 0.0f;
      olab[b * kDets + it] = ok ? (float)cl : -1.0f;
      ms[fi] = kRemoved;  // remove from further selection
    }
    __syncthreads();
  }
}

// ---------------------------------------------------------------------------
extern "C" void kernel_launch(void* const* d_in, const int* in_sizes, int n_in,
                              void* d_out, int out_size, void* d_ws, size_t ws_size,
                              hipStream_t stream) {
  (void)in_sizes; (void)n_in; (void)out_size; (void)ws_size;
  const float* logits = (const float*)d_in[0];   // [B*N, C]
  const float* reg    = (const float*)d_in[1];   // [B*N, 4C]
  const float* props  = (const float*)d_in[2];   // [B*N, 4]

  float* prob   = (float*)d_ws;                          // B*N*C floats
  float* boxes  = prob  + (size_t)kB * kN * kC;          // B*N*C*4 floats
  float* masked = boxes + (size_t)kB * kN * kC * 4;      // B*N*80 floats

  softmax_decode_kernel<<<(kB * kN) / 8, 256, 0, stream>>>(logits, reg, props,
                                                           prob, boxes);

  const size_t smem = 4096      // sorted scores
                    + 4096      // sorted payload
                    + 4 * 4096  // box SoA
                    + 131072    // suppression bitmask
                    + 256;      // counters / pad
  nms_kernel<<<kB * kFG, 1024, smem, stream>>>(prob, boxes, masked);

  topk_kernel<<<kB, 1024, 0, stream>>>(masked, boxes, (float*)d_out);
}